// YonedaEmbedding_9921374454409
// MI455X (gfx1250) — compile-verified
//
#include <hip/hip_runtime.h>
#include <hip/hip_bf16.h>

// Fused row-gather + sigmoid for YonedaEmbedding.
// out[r, :] = sigmoid(logits[idx[r], :]), r = 0..4095, row length V = 16384.
// Pure HBM-bandwidth-bound: ~512 MB moved -> ~22 us at 23.3 TB/s.

#define VDIM 16384
#define TPB  256                 // 8 wave32 waves per block
#define VECS_PER_THREAD (VDIM / (TPB * 4))   // 16 float4 per thread per row

typedef __attribute__((ext_vector_type(4))) float v4f;

__device__ __forceinline__ float fast_sigmoid(float x) {
    // sigmoid(x) = 1 / (1 + 2^(-x * log2(e)))
    // Lowers to v_mul + v_exp_f32 + v_add + v_rcp_f32 (2 TRANS, 2 VALU).
    const float kLog2e = 1.44269504088896340736f;
    float e = __builtin_amdgcn_exp2f(x * -kLog2e);
    return __builtin_amdgcn_rcpf(1.0f + e);
}

__global__ void __launch_bounds__(TPB)
YonedaEmbedding_gather_sigmoid_kernel(const int* __restrict__ idx,
                                      const float* __restrict__ logits,
                                      float* __restrict__ out) {
    // idx[blockIdx.x] is wave-uniform -> scalar (SMEM) load, row base in SGPRs.
    const int row = idx[blockIdx.x];

    const v4f* __restrict__ src =
        (const v4f*)(logits + (size_t)row * (size_t)VDIM);
    v4f* __restrict__ dst =
        (v4f*)(out + (size_t)blockIdx.x * (size_t)VDIM);

    const int t = threadIdx.x;

    // Stage all 16 B128 loads first (64 VGPRs of payload) so the wave has
    // ~16 global_load_b128 in flight before the first s_wait_loadcnt.
    v4f v[VECS_PER_THREAD];
#pragma unroll
    for (int i = 0; i < VECS_PER_THREAD; ++i) {
        v[i] = src[i * TPB + t];          // coalesced: lane-stride 16B
    }

#pragma unroll
    for (int i = 0; i < VECS_PER_THREAD; ++i) {
        v4f r;
        r.x = fast_sigmoid(v[i].x);
        r.y = fast_sigmoid(v[i].y);
        r.z = fast_sigmoid(v[i].z);
        r.w = fast_sigmoid(v[i].w);
        // Output is write-once streaming data: non-temporal store (TH=NT)
        // so the 256 MB output doesn't evict gathered rows from the 192 MB L2
        // (duplicate idx rows then hit L2 instead of re-reading HBM).
        __builtin_nontemporal_store(r, &dst[i * TPB + t]);
    }
}

extern "C" void kernel_launch(void* const* d_in, const int* in_sizes, int n_in,
                              void* d_out, int out_size, void* d_ws, size_t ws_size,
                              hipStream_t stream) {
    const int*   idx    = (const int*)d_in[0];     // [4,1024] flattened = 4096
    const float* logits = (const float*)d_in[1];   // [V, V] f32
    float*       out    = (float*)d_out;           // [4096, V] f32

    const int n_rows = in_sizes[0];                // 4096
    YonedaEmbedding_gather_sigmoid_kernel<<<n_rows, TPB, 0, stream>>>(idx, logits, out);
}